// FTNumericalFeatureEmbedding_55997783605810
// MI455X (gfx1250) — compile-verified
//
#include <hip/hip_runtime.h>

// out[b,f,d] = x[b,f] * W[f,d] + bias[f,d]
// B=16384, F=64, D=192  -> 805 MB of write-once output: pure HBM-store-bound.
// Strategy: per-thread register-resident W4/B4 (staged via gfx1250 async
// global->LDS b128 path), scalar x broadcast loads, nontemporal b128 stores.

typedef float v4f __attribute__((ext_vector_type(4)));
// Exact pointee type of __builtin_amdgcn_global_load_async_to_lds_b128 params
// (from the round-1 diagnostic): gcc-style int4 vector.
typedef int v4i __attribute__((vector_size(4 * sizeof(int))));
typedef __attribute__((address_space(1))) v4i gv4i_t;   // global (AS1)
typedef __attribute__((address_space(3))) v4i lv4i_t;   // LDS    (AS3)
typedef __attribute__((address_space(3))) void lvoid_t;
typedef __attribute__((address_space(3))) char lchar_t;

#define BATCH   16384
#define NFEAT   64
#define EMB     192
#define D4      (EMB / 4)        // 48 float4 lanes per (b,f) row
#define SLOTS   (NFEAT * D4)     // 3072 (f,d4) thread slots
#define TPB     256
#define ROWS    64               // batch rows streamed per block

__global__ __launch_bounds__(TPB) void ftemb_kernel(const float* __restrict__ x,
                                                    const float* __restrict__ W,
                                                    const float* __restrict__ Bv,
                                                    float* __restrict__ out) {
    __shared__ float lds[2 * TPB * 4];   // 8 KB: per-thread W4 + B4 staging

    const int t    = threadIdx.x;
    const int slot = blockIdx.x * TPB + t;      // [0, 3072)
    const int f    = slot / D4;
    const int d4   = slot % D4;

    const float* gW = W  + f * EMB + d4 * 4;
    const float* gB = Bv + f * EMB + d4 * 4;

    // ---- gfx1250 async global->LDS staging of the loop-invariant W / bias ----
#if __has_builtin(__builtin_amdgcn_global_load_async_to_lds_b128)
    __builtin_amdgcn_global_load_async_to_lds_b128(
        (gv4i_t*)(unsigned long long)gW,
        (lv4i_t*)(lvoid_t*)&lds[t * 4], 0, 0);
    __builtin_amdgcn_global_load_async_to_lds_b128(
        (gv4i_t*)(unsigned long long)gB,
        (lv4i_t*)(lvoid_t*)&lds[TPB * 4 + t * 4], 0, 0);
#else
    {
        unsigned ldsW = (unsigned)(unsigned long long)(lchar_t*)&lds[t * 4];
        unsigned ldsB = (unsigned)(unsigned long long)(lchar_t*)&lds[TPB * 4 + t * 4];
        unsigned long long aW = (unsigned long long)gW;
        unsigned long long aB = (unsigned long long)gB;
        asm volatile("global_load_async_to_lds_b128 %0, %2, off\n\t"
                     "global_load_async_to_lds_b128 %1, %3, off"
                     :: "v"(ldsW), "v"(ldsB), "v"(aW), "v"(aB)
                     : "memory");
    }
#endif
#if __has_builtin(__builtin_amdgcn_s_wait_asynccnt)
    __builtin_amdgcn_s_wait_asynccnt(0);
#else
    asm volatile("s_wait_asynccnt 0" ::: "memory");
#endif
    // Each lane reads back only the LDS slots its own async op wrote, so the
    // wave-level ASYNCcnt wait is sufficient (no workgroup barrier needed).
    const v4f w4 = *(const v4f*)&lds[t * 4];
    const v4f b4 = *(const v4f*)&lds[TPB * 4 + t * 4];

    // ---- stream ROWS batch rows: 1 scalar x load + FMA4 + NT b128 store ----
    const int brow   = blockIdx.y * ROWS;
    const float* xp  = x + (unsigned long long)brow * NFEAT + f;
    float*       op  = out + ((unsigned long long)brow * NFEAT + f) * EMB + d4 * 4;

    __builtin_prefetch(xp, 0, 0);   // global_prefetch_b8: warm the x strip

#pragma unroll 4
    for (int i = 0; i < ROWS; ++i) {
        const float xv = *xp;
        v4f o;
        o.x = fmaf(xv, w4.x, b4.x);
        o.y = fmaf(xv, w4.y, b4.y);
        o.z = fmaf(xv, w4.z, b4.z);
        o.w = fmaf(xv, w4.w, b4.w);
        __builtin_nontemporal_store(o, (v4f*)op);   // 805 MB write-once: NT
        xp += NFEAT;
        op += (unsigned long long)NFEAT * EMB;
    }
}

extern "C" void kernel_launch(void* const* d_in, const int* in_sizes, int n_in,
                              void* d_out, int out_size, void* d_ws, size_t ws_size,
                              hipStream_t stream) {
    const float* x  = (const float*)d_in[0];   // [16384, 64]
    const float* W  = (const float*)d_in[1];   // [64, 192]
    const float* b  = (const float*)d_in[2];   // [64, 192]
    float* out      = (float*)d_out;           // [16384, 64, 192]

    dim3 grid(SLOTS / TPB, BATCH / ROWS);      // (12, 256)
    ftemb_kernel<<<grid, dim3(TPB), 0, stream>>>(x, W, b, out);
}